// MRMSNorm_87608742903882
// MI455X (gfx1250) — compile-verified
//
#include <hip/hip_runtime.h>

// fp32 multi-segment RMSNorm, D=4096, SIZES={256,512,1024,2048,4096}.
// Bandwidth-bound: stream x HBM->LDS once via CDNA5 async loads, two cheap
// LDS passes (sum-of-squares, scale+store), nontemporal 128b stores.

typedef float v4f __attribute__((ext_vector_type(4)));
typedef int   v4i __attribute__((ext_vector_type(4)));

// Pointer-to-AS1-int4 (global) and pointer-to-AS3-int4 (LDS).
typedef __attribute__((address_space(1))) v4i* g_v4i_p;
typedef __attribute__((address_space(3))) v4i* l_v4i_p;

__global__ __launch_bounds__(256)
void mrmsnorm_async_kernel(const float* __restrict__ x,
                           const float* __restrict__ scale,
                           float* __restrict__ out)
{
    constexpr int D = 4096;
    __shared__ __align__(16) float tile[D];   // 16 KB row buffer
    __shared__ float bins[16];                // 16 half-block (256-elem) ssq sums
    __shared__ float rrms[16];                // per-half-block rsqrt factor

    const int t = threadIdx.x;
    const long long row = blockIdx.x;
    const float* __restrict__ xrow = x + row * (long long)D;
    float* __restrict__ orow       = out + row * (long long)D;

    // ---- Stage row into LDS: 4 wave-wide async 128b loads per thread. ----
    // float4 index j = i*256 + t -> each wave instruction covers a
    // contiguous 512B global span; LDS offset mirrors global offset.
#pragma unroll
    for (int i = 0; i < 4; ++i) {
        const int j = i * 256 + t;
        // Global flat address == AS1 address numerically: integer round-trip
        // is a safe const-stripping "addrspacecast" for global pointers.
        g_v4i_p gsrc = (g_v4i_p)(unsigned long long)(xrow + 4 * j);
        // LDS needs a real addrspacecast (flat -> LDS offset translation).
        l_v4i_p ldst = (l_v4i_p)(&tile[4 * j]);
        __builtin_amdgcn_global_load_async_to_lds_b128(gsrc, ldst, 0, 0);
    }
#if __has_builtin(__builtin_amdgcn_s_wait_asynccnt)
    __builtin_amdgcn_s_wait_asynccnt(0);
#else
    asm volatile("s_wait_asynccnt 0" ::: "memory");
#endif
    __syncthreads();

    // ---- Pass 1: sum of squares over this thread's 16 contiguous elems. ----
    const v4f* tile4 = (const v4f*)tile;
    float ssq = 0.f;
#pragma unroll
    for (int i = 0; i < 4; ++i) {
        v4f v = tile4[t * 4 + i];
        ssq += v.x * v.x + v.y * v.y + v.z * v.z + v.w * v.w;
    }
    // 16-lane (half-wave32) butterfly reduce: masks 1,2,4,8 keep the two
    // 16-lane halves independent -> one sum per 256-element half-block.
#pragma unroll
    for (int m = 1; m <= 8; m <<= 1)
        ssq += __shfl_xor(ssq, m, 32);
    if ((t & 15) == 0) bins[t >> 4] = ssq;
    __syncthreads();

    // ---- Nested prefixes at the 5 cut points -> rrms per half-block. ----
    if (t < 16) {
        const int n = (t == 0) ? 1 : (t == 1) ? 2 : (t < 4) ? 4 : (t < 8) ? 8 : 16;
        float s = 0.f;
        for (int i = 0; i < n; ++i) s += bins[i];
        rrms[t] = rsqrtf(s * (1.0f / (256.0f * (float)n)) + 1e-6f);
    }
    __syncthreads();

    // ---- Pass 2: out = x * rrms[segment] * scale, streamed NT stores. ----
    const v4f* scale4 = (const v4f*)scale;
#pragma unroll
    for (int i = 0; i < 4; ++i) {
        const int j = i * 256 + t;            // float4 index, lane-contiguous
        const float r = rrms[(4 * j) >> 8];   // half-block id of element 4j
        v4f v  = tile4[j];
        v4f sc = scale4[j];
        v4f o;
        o.x = v.x * r * sc.x;
        o.y = v.y * r * sc.y;
        o.z = v.z * r * sc.z;
        o.w = v.w * r * sc.w;
        __builtin_nontemporal_store(o, (v4f*)(orow + 4 * j));
    }
}

extern "C" void kernel_launch(void* const* d_in, const int* in_sizes, int n_in,
                              void* d_out, int out_size, void* d_ws, size_t ws_size,
                              hipStream_t stream) {
    (void)n_in; (void)d_ws; (void)ws_size; (void)out_size;
    const float* x     = (const float*)d_in[0];
    const float* scale = (const float*)d_in[1];
    float* out         = (float*)d_out;

    const int D = 4096;
    const int rows = in_sizes[0] / D;         // 4 * 4096 = 16384
    mrmsnorm_async_kernel<<<dim3(rows), dim3(256), 0, stream>>>(x, scale, out);
}